// BlockDiffusionDecoder_64819646431740
// MI455X (gfx1250) — compile-verified
//
#include <hip/hip_runtime.h>

typedef float v2f __attribute__((ext_vector_type(2)));
typedef float v8f __attribute__((ext_vector_type(8)));
typedef int   v4i __attribute__((ext_vector_type(4)));

#define B_    2
#define H_    16
#define S_    2048
#define D_    64
#define BLK   64
#define LSTR  68          // LDS row stride (floats): 16B-aligned rows, conflict-free reads
#define SCALE 0.125f      // 1/sqrt(64)

// ---- CDNA5 async global->LDS copy (ASYNCcnt-tracked), with fallback ----
#if defined(__has_builtin)
#  if __has_builtin(__builtin_amdgcn_global_load_async_to_lds_b128)
#    define HAVE_ASYNC_LDS 1
#  else
#    define HAVE_ASYNC_LDS 0
#  endif
#  if __has_builtin(__builtin_amdgcn_s_wait_asynccnt)
#    define HAVE_WAIT_ASYNC 1
#  else
#    define HAVE_WAIT_ASYNC 0
#  endif
#else
#  define HAVE_ASYNC_LDS 0
#  define HAVE_WAIT_ASYNC 0
#endif

__device__ __forceinline__ void async_copy_b128(const float* g, float* l) {
#if HAVE_ASYNC_LDS
    // param types (from hipcc diagnostic): (v4i* global, v4i* lds, imm offset, imm cpol)
    __builtin_amdgcn_global_load_async_to_lds_b128((v4i*)g, (v4i*)l, /*offset=*/0, /*cpol=*/0);
#else
    *(float4*)l = *(const float4*)g;
#endif
}

__device__ __forceinline__ void wait_async_copies() {
#if HAVE_ASYNC_LDS
#  if HAVE_WAIT_ASYNC
    __builtin_amdgcn_s_wait_asynccnt(0);
#  else
    asm volatile("s_wait_asynccnt 0x0" ::: "memory");
#  endif
#endif
}

__global__ __launch_bounds__(128)
void fa_block_causal_f32(const float* __restrict__ q,
                         const float* __restrict__ k,
                         const float* __restrict__ v,
                         float* __restrict__ out)
{
    __shared__ __align__(16) float Ksh[2][BLK * LSTR];
    __shared__ __align__(16) float Vsh[2][BLK * LSTR];
    __shared__ __align__(16) float Psh[BLK * LSTR];

    const int tid  = threadIdx.x;
    const int wave = tid >> 5;
    const int lane = tid & 31;
    const int l16  = lane & 15;
    const int hi   = lane >> 4;            // which half of the wave

    const int qb = blockIdx.x;             // query block index [0,32)
    const int bh = blockIdx.y;             // b*H + h
    const size_t base = (size_t)bh * S_ * D_;

    // this thread's 8 chunk coordinates for cooperative 64x64 f32 staging
    // (idx = tid + j*128 -> row = idx>>4, col4 = (idx&15)*4; 16B chunks)
    const int c_row0 = tid >> 4;           // rows this thread covers: c_row0 + 8*j
    const int c_col  = (tid & 15) << 2;

    // ---- Q A-fragments (16x4 layout per k-step), pre-scaled by 1/sqrt(D) ----
    const int qrow = qb * BLK + wave * 16 + l16;
    const float* qptr = q + base + (size_t)qrow * D_;
    v2f qa[16];
#pragma unroll
    for (int ks = 0; ks < 16; ++ks) {
        const int c = 4 * ks + 2 * hi;
        qa[ks].x = qptr[c]     * SCALE;
        qa[ks].y = qptr[c + 1] * SCALE;
    }

    v8f o0 = {}, o1 = {}, o2 = {}, o3 = {};
    float m_i[8], l_i[8];
#pragma unroll
    for (int r = 0; r < 8; ++r) { m_i[r] = -1e30f; l_i[r] = 0.0f; }

    // ---- prologue: kick off async staging of key-block 0 into buffer 0 ----
    {
        const float* kg = k + base;
        const float* vg = v + base;
#pragma unroll
        for (int j = 0; j < 8; ++j) {
            const int row = c_row0 + j * 8;
            async_copy_b128(kg + row * D_ + c_col, &Ksh[0][row * LSTR + c_col]);
            async_copy_b128(vg + row * D_ + c_col, &Vsh[0][row * LSTR + c_col]);
        }
    }

    for (int kb = 0; kb <= qb; ++kb) {
        const int buf = kb & 1;

        wait_async_copies();               // this wave's copies into `buf` done
        __syncthreads();                   // all waves' copies done; prev reads of buf^1 retired

        // ---- overlap: start async staging of next key block into buf^1 ----
        if (kb < qb) {
            const float* kg = k + base + (size_t)((kb + 1) * BLK) * D_;
            const float* vg = v + base + (size_t)((kb + 1) * BLK) * D_;
#pragma unroll
            for (int j = 0; j < 8; ++j) {
                const int row = c_row0 + j * 8;
                async_copy_b128(kg + row * D_ + c_col, &Ksh[buf ^ 1][row * LSTR + c_col]);
                async_copy_b128(vg + row * D_ + c_col, &Vsh[buf ^ 1][row * LSTR + c_col]);
            }
        }

        const float* Kb = Ksh[buf];
        const float* Vb = Vsh[buf];

        // ---- S = (Q * 1/sqrt(D)) . K^T : 4 tiles of 16x16, fp32 WMMA ----
        v8f s0 = {}, s1 = {}, s2 = {}, s3 = {};
#pragma unroll
        for (int ks = 0; ks < 16; ++ks) {
            const int c = 4 * ks + 2 * hi;
            const v2f b0 = *(const v2f*)(&Kb[( 0 + l16) * LSTR + c]);
            const v2f b1 = *(const v2f*)(&Kb[(16 + l16) * LSTR + c]);
            const v2f b2 = *(const v2f*)(&Kb[(32 + l16) * LSTR + c]);
            const v2f b3 = *(const v2f*)(&Kb[(48 + l16) * LSTR + c]);
            s0 = __builtin_amdgcn_wmma_f32_16x16x4_f32(false, qa[ks], false, b0, (short)0, s0, false, false);
            s1 = __builtin_amdgcn_wmma_f32_16x16x4_f32(false, qa[ks], false, b1, (short)0, s1, false, false);
            s2 = __builtin_amdgcn_wmma_f32_16x16x4_f32(false, qa[ks], false, b2, (short)0, s2, false, false);
            s3 = __builtin_amdgcn_wmma_f32_16x16x4_f32(false, qa[ks], false, b3, (short)0, s3, false, false);
        }

        // ---- online softmax: all 4 tiles share the row<->(vgpr,half) mapping ----
#pragma unroll
        for (int r = 0; r < 8; ++r) {
            float t = fmaxf(fmaxf(s0[r], s1[r]), fmaxf(s2[r], s3[r]));
            t = fmaxf(t, __shfl_xor(t, 1, 32));
            t = fmaxf(t, __shfl_xor(t, 2, 32));
            t = fmaxf(t, __shfl_xor(t, 4, 32));
            t = fmaxf(t, __shfl_xor(t, 8, 32));
            const float mr = fmaxf(m_i[r], t);

            s0[r] = __expf(s0[r] - mr);
            s1[r] = __expf(s1[r] - mr);
            s2[r] = __expf(s2[r] - mr);
            s3[r] = __expf(s3[r] - mr);
            float ps = s0[r] + s1[r] + s2[r] + s3[r];
            ps += __shfl_xor(ps, 1, 32);
            ps += __shfl_xor(ps, 2, 32);
            ps += __shfl_xor(ps, 4, 32);
            ps += __shfl_xor(ps, 8, 32);

            const float alpha = __expf(m_i[r] - mr);
            l_i[r] = l_i[r] * alpha + ps;
            m_i[r] = mr;
            o0[r] *= alpha; o1[r] *= alpha; o2[r] *= alpha; o3[r] *= alpha;
        }

        // ---- transpose P (C/D layout -> A layout) via wave-private LDS tile ----
#pragma unroll
        for (int r = 0; r < 8; ++r) {
            const int prow = wave * 16 + r + 8 * hi;
            Psh[prow * LSTR +  0 + l16] = s0[r];
            Psh[prow * LSTR + 16 + l16] = s1[r];
            Psh[prow * LSTR + 32 + l16] = s2[r];
            Psh[prow * LSTR + 48 + l16] = s3[r];
        }

        // ---- O += P . V ----
        const float* pbase = &Psh[(wave * 16 + l16) * LSTR];
#pragma unroll
        for (int ks = 0; ks < 16; ++ks) {
            const int c = 4 * ks + 2 * hi;          // key-row base within block
            const v2f a = *(const v2f*)(pbase + c);
            v2f b0, b1, b2, b3;
            b0.x = Vb[ c      * LSTR +  0 + l16];
            b0.y = Vb[(c + 1) * LSTR +  0 + l16];
            b1.x = Vb[ c      * LSTR + 16 + l16];
            b1.y = Vb[(c + 1) * LSTR + 16 + l16];
            b2.x = Vb[ c      * LSTR + 32 + l16];
            b2.y = Vb[(c + 1) * LSTR + 32 + l16];
            b3.x = Vb[ c      * LSTR + 48 + l16];
            b3.y = Vb[(c + 1) * LSTR + 48 + l16];
            o0 = __builtin_amdgcn_wmma_f32_16x16x4_f32(false, a, false, b0, (short)0, o0, false, false);
            o1 = __builtin_amdgcn_wmma_f32_16x16x4_f32(false, a, false, b1, (short)0, o1, false, false);
            o2 = __builtin_amdgcn_wmma_f32_16x16x4_f32(false, a, false, b2, (short)0, o2, false, false);
            o3 = __builtin_amdgcn_wmma_f32_16x16x4_f32(false, a, false, b3, (short)0, o3, false, false);
        }
    }

    // ---- epilogue: O / l, store (C/D layout -> row-major global) ----
    float* optr = out + base + (size_t)(qb * BLK + wave * 16) * D_;
#pragma unroll
    for (int r = 0; r < 8; ++r) {
        const float inv = 1.0f / l_i[r];
        const int row = r + 8 * hi;
        optr[row * D_ +  0 + l16] = o0[r] * inv;
        optr[row * D_ + 16 + l16] = o1[r] * inv;
        optr[row * D_ + 32 + l16] = o2[r] * inv;
        optr[row * D_ + 48 + l16] = o3[r] * inv;
    }
}

extern "C" void kernel_launch(void* const* d_in, const int* in_sizes, int n_in,
                              void* d_out, int out_size, void* d_ws, size_t ws_size,
                              hipStream_t stream) {
    (void)in_sizes; (void)n_in; (void)out_size; (void)d_ws; (void)ws_size;
    const float* q = (const float*)d_in[0];
    const float* k = (const float*)d_in[1];
    const float* v = (const float*)d_in[2];
    float* out = (float*)d_out;
    dim3 grid(S_ / BLK, B_ * H_);
    fa_block_causal_f32<<<grid, 128, 0, stream>>>(q, k, v, out);
}